// Embedding2Score_73813307949584
// MI455X (gfx1250) — compile-verified
//
#include <hip/hip_runtime.h>
#include <hip/hip_bf16.h>

typedef __attribute__((ext_vector_type(2))) float v2f;
typedef __attribute__((ext_vector_type(8))) float v8f;

#define HID 128
#define NSEG 2048

// ---------------------------------------------------------------------------
// K1: find last index of each (sorted, contiguous, non-empty) segment
// ---------------------------------------------------------------------------
__global__ void __launch_bounds__(256)
seg_last_kernel(const int* __restrict__ batch, int* __restrict__ last_idx, int N) {
    int i = blockIdx.x * blockDim.x + threadIdx.x;
    if (i >= N) return;
    if (i == N - 1 || batch[i] != batch[i + 1]) {
        last_idx[batch[i]] = i;
    }
}

// ---------------------------------------------------------------------------
// K2: v_n[s] = x[last_idx[s]] ; also zero s_g accumulator
// ---------------------------------------------------------------------------
__global__ void __launch_bounds__(128)
gather_vn_kernel(const float* __restrict__ x, const int* __restrict__ last_idx,
                 float* __restrict__ v_n, float* __restrict__ s_g) {
    int s = blockIdx.x;
    int t = threadIdx.x;
    v_n[(long)s * HID + t] = x[(long)last_idx[s] * HID + t];
    s_g[(long)s * HID + t] = 0.0f;
}

// ---------------------------------------------------------------------------
// Generic WMMA f32 GEMM (16x16 tile / wave) for the small GEMMs:
//   D[m][n] = sum_k A[m][k] * Bt[n][k]  (+ bias[n])
// ---------------------------------------------------------------------------
__global__ void __launch_bounds__(256)
wmma_gemm_abT(const float* __restrict__ A, const float* __restrict__ Bt,
              const float* __restrict__ bias, float* __restrict__ D,
              int M, int N, int K, int lda, int ldb, int ldd) {
    const int lane = threadIdx.x & 31;
    const int wave = threadIdx.x >> 5;
    const int tilesN = (N + 15) >> 4;
    const long totalTiles = (long)((M + 15) >> 4) * tilesN;
    long tile = (long)blockIdx.x * 8 + wave;           // wave-uniform
    if (tile >= totalTiles) return;                    // uniform exit: EXEC stays full

    const int mt = (int)(tile / tilesN);
    const int nt = (int)(tile % tilesN);
    const int m0 = mt << 4;
    const int n0 = nt << 4;

    const int row  = lane & 15;   // A: M index / B: N index within tile
    const int kh   = lane >> 4;   // K-half select (0 -> k,k+1 ; 1 -> k+2,k+3)

    int am = m0 + row; if (am >= M) am = M - 1;
    int bn = n0 + row; if (bn >= N) bn = N - 1;
    const float* aptr = A  + (long)am * lda + 2 * kh;
    const float* bptr = Bt + (long)bn * ldb + 2 * kh;

    v8f acc = (v8f){0.f,0.f,0.f,0.f,0.f,0.f,0.f,0.f};
#pragma unroll 8
    for (int k0 = 0; k0 < K; k0 += 4) {
        v2f a = *(const v2f*)(aptr + k0);
        v2f b = *(const v2f*)(bptr + k0);
        acc = __builtin_amdgcn_wmma_f32_16x16x4_f32(
            false, a, false, b, (short)0, acc, false, false);
    }

    const int n = n0 + (lane & 15);
    if (n < N) {
        const float bv = bias ? bias[n] : 0.0f;
        const int mbase = m0 + 8 * kh;
#pragma unroll
        for (int v = 0; v < 8; ++v) {
            int m = mbase + v;
            if (m < M) D[(long)m * ldd + n] = acc[v] + bv;
        }
    }
}

// ---------------------------------------------------------------------------
// K7: register-blocked WMMA GEMM for the dominant op  out = A @ Bt^T.
// One wave32 computes a 64x64 output block as 4x4 subtiles of 16x16
// (16 v8f accumulators = 128 VGPRs). Per k-step: 4 A-loads + 4 B-loads
// feed 16 v_wmma_f32_16x16x4_f32 -> 0.5 loads/wmma (4x less L2 traffic
// than the 16x16 kernel). M must be a multiple of 64; N ragged edge is
// handled by clamped B loads + masked stores.
// ---------------------------------------------------------------------------
__global__ void __launch_bounds__(256)
wmma_gemm_abT_64x64(const float* __restrict__ A, const float* __restrict__ Bt,
                    float* __restrict__ D, int M, int N, int K,
                    int lda, int ldb, int ldd) {
    const int lane = threadIdx.x & 31;
    const int wave = threadIdx.x >> 5;
    const int blocksN = (N + 63) >> 6;
    const long totalBlk = (long)(M >> 6) * blocksN;
    long blk = (long)blockIdx.x * 8 + wave;            // wave-uniform
    if (blk >= totalBlk) return;                       // uniform exit

    // consecutive waves in a workgroup -> consecutive N blocks, same M block
    const int mb = (int)(blk / blocksN);
    const int nb = (int)(blk % blocksN);
    const int m0 = mb << 6;
    const int n0 = nb << 6;

    const int row = lane & 15;
    const int kh  = lane >> 4;

    const float* aptr[4];
    const float* bptr[4];
#pragma unroll
    for (int i = 0; i < 4; ++i) {
        aptr[i] = A + (long)(m0 + 16 * i + row) * lda + 2 * kh;
        int bn = n0 + 16 * i + row; if (bn >= N) bn = N - 1;   // clamp ragged edge
        bptr[i] = Bt + (long)bn * ldb + 2 * kh;
    }

    v8f acc[4][4];
#pragma unroll
    for (int i = 0; i < 4; ++i)
#pragma unroll
        for (int j = 0; j < 4; ++j)
            acc[i][j] = (v8f){0.f,0.f,0.f,0.f,0.f,0.f,0.f,0.f};

#pragma unroll 2
    for (int k0 = 0; k0 < K; k0 += 4) {
        v2f a[4], b[4];
#pragma unroll
        for (int i = 0; i < 4; ++i) a[i] = *(const v2f*)(aptr[i] + k0);
#pragma unroll
        for (int j = 0; j < 4; ++j) b[j] = *(const v2f*)(bptr[j] + k0);
#pragma unroll
        for (int i = 0; i < 4; ++i)
#pragma unroll
            for (int j = 0; j < 4; ++j)
                acc[i][j] = __builtin_amdgcn_wmma_f32_16x16x4_f32(
                    false, a[i], false, b[j], (short)0, acc[i][j], false, false);
    }

    // store: subtile (i,j): D[m0+16i + 8*kh + v][n0+16j + (lane&15)]
    const int ncol = lane & 15;
#pragma unroll
    for (int j = 0; j < 4; ++j) {
        const int n = n0 + 16 * j + ncol;
        if (n < N) {
#pragma unroll
            for (int i = 0; i < 4; ++i) {
                float* drow = D + (long)(m0 + 16 * i + 8 * kh) * ldd + n;
#pragma unroll
                for (int v = 0; v < 8; ++v) {
                    drow[(long)v * ldd] = acc[i][j][v];
                }
            }
        }
    }
}

// ---------------------------------------------------------------------------
// K4: fused attention. One wave32 = 16 nodes, full 128 h-dims:
//   pre  = x_tile @ W2^T            (8 x V_WMMA_F32_16X16X4_F32 accumulators)
//   h    = sigmoid(pre + w1_vn[batch] + W2_b)
//   alpha= h . q_w + q_b            (half-wave shfl_xor reduction)
//   s_g[batch] += alpha * x         (global_atomic_add_f32)
// ---------------------------------------------------------------------------
__global__ void __launch_bounds__(256)
node_attn_kernel(const float* __restrict__ x, const float* __restrict__ W2w,
                 const float* __restrict__ W2b, const float* __restrict__ w1vn,
                 const float* __restrict__ qw, const float* __restrict__ qb,
                 const int* __restrict__ batch, float* __restrict__ sg, int Nnodes) {
    const int lane = threadIdx.x & 31;
    const int wave = threadIdx.x >> 5;
    const int tile = blockIdx.x * 8 + wave;
    const int m0 = tile << 4;
    if (m0 >= Nnodes) return;                          // wave-uniform

    const int row = lane & 15;
    const int kh  = lane >> 4;
    const float* aptr  = x   + (long)(m0 + row) * HID + 2 * kh;
    const float* bbase = W2w + (long)row * HID + 2 * kh;

    v8f acc[8];
#pragma unroll
    for (int j = 0; j < 8; ++j) acc[j] = (v8f){0.f,0.f,0.f,0.f,0.f,0.f,0.f,0.f};

    for (int k0 = 0; k0 < HID; k0 += 4) {
        v2f a = *(const v2f*)(aptr + k0);
#pragma unroll
        for (int j = 0; j < 8; ++j) {                  // h dims 16j..16j+15
            v2f b = *(const v2f*)(bbase + (long)j * 16 * HID + k0);
            acc[j] = __builtin_amdgcn_wmma_f32_16x16x4_f32(
                false, a, false, b, (short)0, acc[j], false, false);
        }
    }

    const int ncol = lane & 15;
    int   bseg[8];
    float pal[8];
#pragma unroll
    for (int v = 0; v < 8; ++v) {
        bseg[v] = batch[m0 + 8 * kh + v];
        pal[v] = 0.0f;
    }
#pragma unroll
    for (int j = 0; j < 8; ++j) {
        const int h   = 16 * j + ncol;
        const float qh  = qw[h];
        const float b2h = W2b[h];
#pragma unroll
        for (int v = 0; v < 8; ++v) {
            float pre = acc[j][v] + w1vn[(long)bseg[v] * HID + h] + b2h;
            float hv  = 1.0f / (1.0f + __expf(-pre));  // sigmoid
            pal[v] += qh * hv;
        }
    }
    const float qbv = qb[0];
#pragma unroll
    for (int v = 0; v < 8; ++v) {
        float al = pal[v];
        for (int off = 8; off >= 1; off >>= 1) al += __shfl_xor(al, off, 16);
        al += qbv;
        const int m = m0 + 8 * kh + v;
        const float* xr = x + (long)m * HID;
        float* sgr = sg + (long)bseg[v] * HID;
#pragma unroll
        for (int c = 0; c < 8; ++c) {
            int hh = ncol + 16 * c;
            atomicAdd(&sgr[hh], al * xr[hh]);
        }
    }
}

// ---------------------------------------------------------------------------
// K5: cat[s] = [ v_n[s] ; s_g[s] ]
// ---------------------------------------------------------------------------
__global__ void __launch_bounds__(256)
concat_kernel(const float* __restrict__ v_n, const float* __restrict__ s_g,
              float* __restrict__ cat) {
    int s = blockIdx.x;
    int t = threadIdx.x;
    cat[(long)s * (2 * HID) + t] = (t < HID) ? v_n[(long)s * HID + t]
                                             : s_g[(long)s * HID + (t - HID)];
}

// ---------------------------------------------------------------------------
extern "C" void kernel_launch(void* const* d_in, const int* in_sizes, int n_in,
                              void* d_out, int out_size, void* d_ws, size_t ws_size,
                              hipStream_t stream) {
    const float* x    = (const float*)d_in[0];   // [N,128]
    const float* E    = (const float*)d_in[1];   // [V,128]
    const float* W1w  = (const float*)d_in[2];   // [128,128]
    const float* W1b  = (const float*)d_in[3];   // [128]
    const float* W2w  = (const float*)d_in[4];   // [128,128]
    const float* W2b  = (const float*)d_in[5];   // [128]
    const float* qw   = (const float*)d_in[6];   // [1,128]
    const float* qb   = (const float*)d_in[7];   // [1]
    const float* W3w  = (const float*)d_in[8];   // [128,256]
    const float* W3b  = (const float*)d_in[9];   // [128]
    const int*   batch= (const int*)d_in[10];    // [N]
    float* out = (float*)d_out;                  // [B, V-1]

    const int N  = in_sizes[0] / HID;            // 102400
    const int V  = in_sizes[1] / HID;            // 100000
    const int B  = NSEG;                         // 2048
    const int Nv = V - 1;                        // 99999

    // workspace carve-out (~6.3 MB)
    char* w = (char*)d_ws;
    int*   last_idx = (int*)w;   w += ((B * sizeof(int) + 255) / 256) * 256;
    float* v_n  = (float*)w;     w += (size_t)B * HID * sizeof(float);
    float* w1vn = (float*)w;     w += (size_t)B * HID * sizeof(float);
    float* s_g  = (float*)w;     w += (size_t)B * HID * sizeof(float);
    float* cat  = (float*)w;     w += (size_t)B * 2 * HID * sizeof(float);
    float* s_h  = (float*)w;     w += (size_t)B * HID * sizeof(float);

    // K1: segment boundaries
    seg_last_kernel<<<(N + 255) / 256, 256, 0, stream>>>(batch, last_idx, N);

    // K2: gather v_n, zero s_g
    gather_vn_kernel<<<B, 128, 0, stream>>>(x, last_idx, v_n, s_g);

    // K3: w1_vn = v_n @ W1^T + W1_b   (2048x128x128)
    {
        long tiles = (long)(B / 16) * (HID / 16);
        wmma_gemm_abT<<<(int)((tiles + 7) / 8), 256, 0, stream>>>(
            v_n, W1w, W1b, w1vn, B, HID, HID, HID, HID, HID);
    }

    // K4: fused h / alpha / segment-sum  (102400x128x128 + epilogue)
    {
        int tiles = N / 16;                       // 6400
        node_attn_kernel<<<(tiles + 7) / 8, 256, 0, stream>>>(
            x, W2w, W2b, w1vn, qw, qb, batch, s_g, N);
    }

    // K5: cat = [v_n ; s_g]
    concat_kernel<<<B, 2 * HID, 0, stream>>>(v_n, s_g, cat);

    // K6: s_h = cat @ W3^T + W3_b     (2048x128x256)
    {
        long tiles = (long)(B / 16) * (HID / 16);
        wmma_gemm_abT<<<(int)((tiles + 7) / 8), 256, 0, stream>>>(
            cat, W3w, W3b, s_h, B, HID, 2 * HID, 2 * HID, 2 * HID, HID);
    }

    // K7: out = s_h @ E[1:]^T  (2048x99999x128, 52.4 GFLOP) — 64x64 blocked
    {
        long blks = (long)(B / 64) * ((Nv + 63) / 64);   // 32 * 1563 = 50016
        wmma_gemm_abT_64x64<<<(int)((blks + 7) / 8), 256, 0, stream>>>(
            s_h, E + HID, out, B, Nv, HID, HID, HID, Nv);
    }
}